// MMLULlamaHead_7550552506703
// MI455X (gfx1250) — compile-verified
//
#include <hip/hip_runtime.h>
#include <math.h>

// Problem constants (match reference)
#define D_MODEL 4096
#define N_OPT   4
#define RMS_EPS 1e-5f

#define BLOCK   256
#define WAVES   (BLOCK / 32)          // 8 waves (wave32)
#define K_SLICE (D_MODEL / WAVES)     // 512 K per wave

typedef float v2f __attribute__((ext_vector_type(2)));
typedef float v8f __attribute__((ext_vector_type(8)));

// Fully fused: RMSNorm(last token) -> 4 gathered dot products (f32 WMMA) -> softmax.
// Single workgroup; total global traffic ~80KB, so this is latency-bound by design.
__global__ __launch_bounds__(BLOCK)
void mmlu_llama_head_kernel(const float* __restrict__ x,
                            const float* __restrict__ norm_w,
                            const float* __restrict__ head_w,
                            const int*   __restrict__ opt_ids,
                            float* __restrict__ out,
                            int seq)
{
    __shared__ __align__(16) float h_lds[D_MODEL];   // normalized hidden state (16 KB)
    __shared__ float red[BLOCK];                     // reduction scratch
    __shared__ float part[WAVES][N_OPT];             // per-wave partial dots

    const int tid  = threadIdx.x;
    const int lane = tid & 31;
    const int wave = tid >> 5;

    const float* xrow = x + (size_t)(seq - 1) * (size_t)D_MODEL;

    // ---- Warm the caches with the 4 live head rows while we do the RMS pass.
    // 256 threads * 16 floats (64B) = one full 16KB row per option.
    {
        #pragma unroll
        for (int i = 0; i < N_OPT; ++i) {
            const float* pr = head_w + (size_t)opt_ids[i] * (size_t)D_MODEL;
            __builtin_prefetch(pr + tid * 16, 0, 0);   // -> global_prefetch_b8
        }
    }

    // ---- Phase 1: sum of squares of the last token (coalesced, 16 elems/thread)
    float ss = 0.0f;
    #pragma unroll
    for (int j = tid; j < D_MODEL; j += BLOCK) {
        float v = xrow[j];
        ss += v * v;
    }
    red[tid] = ss;
    __syncthreads();
    #pragma unroll
    for (int s = BLOCK / 2; s > 0; s >>= 1) {
        if (tid < s) red[tid] += red[tid + s];
        __syncthreads();
    }
    const float rinv = rsqrtf(red[0] * (1.0f / (float)D_MODEL) + RMS_EPS);

    // h = x * rinv * w  -> LDS
    for (int j = tid; j < D_MODEL; j += BLOCK) {
        h_lds[j] = xrow[j] * rinv * norm_w[j];
    }
    __syncthreads();

    // ---- Phase 2: 4 dot products via V_WMMA_F32_16X16X4_F32, K split over 8 waves.
    // A (16x4): rows 0..3 = head rows for the 4 options, rows 4..15 = 0.
    //   A VGPR layout: lanes 0-15 -> M=lane, K=0 (v0) / K=1 (v1);
    //                  lanes 16-31 -> M=lane-16, K=2 (v0) / K=3 (v1).
    // B (4x16): broadcast h[k] across all 16 columns.
    const int  mrow = lane & 15;
    const int  koff = (lane < 16) ? 0 : 2;
    const bool live = (mrow < N_OPT);
    // Dead lanes alias vocab row 0 -> loads are valid for ALL lanes (no exec
    // branching); the 'live' mask is applied AFTER the load as a v_cndmask.
    const int  rid  = live ? opt_ids[mrow] : 0;
    const float* wr = head_w + (size_t)rid * (size_t)D_MODEL;

    v8f c = {0.f, 0.f, 0.f, 0.f, 0.f, 0.f, 0.f, 0.f};
    const int kbase = wave * K_SLICE;

    #pragma unroll 8
    for (int it = 0; it < K_SLICE / 4; ++it) {
        const int kb = kbase + it * 4 + koff;     // kb is even -> 8B aligned
        v2f av = *(const v2f*)(wr + kb);          // unconditional global_load_b64
        v2f bv = *(const v2f*)(h_lds + kb);       // ds_load_b64
        v2f a;
        a.x = live ? av.x : 0.0f;                 // v_cndmask (exact zero)
        a.y = live ? av.y : 0.0f;
        // 8 args: (neg_a, A, neg_b, B, c_mod, C, reuse_a, reuse_b)
        c = __builtin_amdgcn_wmma_f32_16x16x4_f32(
                false, a, false, bv, (short)0, c, false, false);
    }

    // D layout: lane 0 holds D[m][0] in c[m] for m = 0..7  -> options live in c[0..3]
    if (lane == 0) {
        part[wave][0] = c[0];
        part[wave][1] = c[1];
        part[wave][2] = c[2];
        part[wave][3] = c[3];
    }
    __syncthreads();

    // ---- Phase 3: deterministic cross-wave sum + softmax on thread 0
    if (tid == 0) {
        float logit[N_OPT];
        #pragma unroll
        for (int i = 0; i < N_OPT; ++i) {
            float acc = 0.0f;
            #pragma unroll
            for (int w = 0; w < WAVES; ++w) acc += part[w][i];
            logit[i] = acc;
        }
        float mx = logit[0];
        #pragma unroll
        for (int i = 1; i < N_OPT; ++i) mx = fmaxf(mx, logit[i]);
        float e[N_OPT];
        float es = 0.0f;
        #pragma unroll
        for (int i = 0; i < N_OPT; ++i) { e[i] = expf(logit[i] - mx); es += e[i]; }
        const float inv = 1.0f / es;
        #pragma unroll
        for (int i = 0; i < N_OPT; ++i) out[i] = e[i] * inv;
    }
}

extern "C" void kernel_launch(void* const* d_in, const int* in_sizes, int n_in,
                              void* d_out, int out_size, void* d_ws, size_t ws_size,
                              hipStream_t stream) {
    const float* x      = (const float*)d_in[0];   // [SEQ, D_MODEL] f32
    const float* norm_w = (const float*)d_in[1];   // [D_MODEL] f32
    const float* head_w = (const float*)d_in[2];   // [VOCAB, D_MODEL] f32
    const int*   optids = (const int*)  d_in[3];   // [4] i32

    float* out = (float*)d_out;                    // [4] f32
    const int seq = in_sizes[0] / D_MODEL;

    mmlu_llama_head_kernel<<<1, BLOCK, 0, stream>>>(x, norm_w, head_w, optids, out, seq);
}